// EstimateRebuildModule_60284160966790
// MI455X (gfx1250) — compile-verified
//
#include <hip/hip_runtime.h>
#include <hip/hip_bf16.h>
#include <math.h>

typedef __attribute__((ext_vector_type(2))) float v2f;
typedef __attribute__((ext_vector_type(8))) float v8f;

#define BATCH 64
#define HDIM 256
#define WDIM 256
#define HW (HDIM * WDIM)
#define NKPT 68
#define TILE 2048
#define PTILE (TILE + 16)             // padded LDS plane stride (banks shift by 16/plane)
#define TILES_PER_BATCH (HW / TILE)   // 32

// ---------------------------------------------------------------------------
// Kernel 1: per-batch Umeyama (similarity transform) on 68 keypoints.
// One thread per batch; writes the 4x16 WMMA B-operand (rows k=0..3,
// cols d=0..3 used; B[c][d]=sR[d][c], B[3][d]=t[d]) to ws (16 floats/batch).
// ---------------------------------------------------------------------------
__global__ void umeyama_kernel(const float* __restrict__ Off,
                               const float* __restrict__ Pos,
                               const float* __restrict__ Mean,
                               const int*   __restrict__ uv,
                               float*       __restrict__ wsB)
{
    int b = blockIdx.x * blockDim.x + threadIdx.x;
    if (b >= BATCH) return;

    const float rev[3] = {1.f, -1.f, 1.f};
    const float* ob = Off + (size_t)b * 3 * HW;
    const float* pb = Pos + (size_t)b * 3 * HW;

    float ss[3] = {0.f, 0.f, 0.f};
    float sd[3] = {0.f, 0.f, 0.f};
    float ssq = 0.f;
    float C[3][3] = {{0,0,0},{0,0,0},{0,0,0}};  // sum dst_i * src_i^T

    for (int k = 0; k < NKPT; ++k) {
        int p = uv[2*k] * WDIM + uv[2*k + 1];
        float s[3], d[3];
        #pragma unroll
        for (int c = 0; c < 3; ++c) {
            s[c] = rev[c] * (ob[(size_t)c * HW + p] * 4.0f + Mean[(size_t)c * HW + p]);
            d[c] = pb[(size_t)c * HW + p];
        }
        #pragma unroll
        for (int c = 0; c < 3; ++c) { ss[c] += s[c]; sd[c] += d[c]; }
        ssq += s[0]*s[0] + s[1]*s[1] + s[2]*s[2];
        #pragma unroll
        for (int i = 0; i < 3; ++i)
            #pragma unroll
            for (int j = 0; j < 3; ++j)
                C[i][j] += d[i] * s[j];
    }

    const float inv_n = 1.0f / (float)NKPT;
    float mus[3], mud[3];
    #pragma unroll
    for (int c = 0; c < 3; ++c) { mus[c] = ss[c] * inv_n; mud[c] = sd[c] * inv_n; }

    float A[3][3];
    #pragma unroll
    for (int i = 0; i < 3; ++i)
        #pragma unroll
        for (int j = 0; j < 3; ++j)
            A[i][j] = C[i][j] * inv_n - mud[i] * mus[j];

    float var_s = ssq * inv_n - (mus[0]*mus[0] + mus[1]*mus[1] + mus[2]*mus[2]);

    float detA = A[0][0]*(A[1][1]*A[2][2] - A[1][2]*A[2][1])
               - A[0][1]*(A[1][0]*A[2][2] - A[1][2]*A[2][0])
               + A[0][2]*(A[1][0]*A[2][1] - A[1][1]*A[2][0]);

    // M = A^T A (symmetric), Jacobi eigensolve -> V, eigenvalues
    float M[3][3];
    #pragma unroll
    for (int i = 0; i < 3; ++i)
        #pragma unroll
        for (int j = 0; j < 3; ++j)
            M[i][j] = A[0][i]*A[0][j] + A[1][i]*A[1][j] + A[2][i]*A[2][j];

    float V[3][3] = {{1,0,0},{0,1,0},{0,0,1}};
    const int PP[3] = {0, 0, 1};
    const int QQ[3] = {1, 2, 2};
    for (int sweep = 0; sweep < 8; ++sweep) {
        for (int r = 0; r < 3; ++r) {
            int p = PP[r], q = QQ[r];
            float apq = M[p][q];
            if (fabsf(apq) > 1e-20f) {
                float tau = (M[q][q] - M[p][p]) / (2.0f * apq);
                float tt  = ((tau >= 0.f) ? 1.f : -1.f) / (fabsf(tau) + sqrtf(1.f + tau*tau));
                float cc  = rsqrtf(1.f + tt*tt);
                float sn  = tt * cc;
                #pragma unroll
                for (int k = 0; k < 3; ++k) {
                    float mkp = M[k][p], mkq = M[k][q];
                    M[k][p] = cc*mkp - sn*mkq;
                    M[k][q] = sn*mkp + cc*mkq;
                }
                #pragma unroll
                for (int k = 0; k < 3; ++k) {
                    float mpk = M[p][k], mqk = M[q][k];
                    M[p][k] = cc*mpk - sn*mqk;
                    M[q][k] = sn*mpk + cc*mqk;
                }
                #pragma unroll
                for (int k = 0; k < 3; ++k) {
                    float vkp = V[k][p], vkq = V[k][q];
                    V[k][p] = cc*vkp - sn*vkq;
                    V[k][q] = sn*vkp + cc*vkq;
                }
            }
        }
    }

    float sig[3];
    #pragma unroll
    for (int i = 0; i < 3; ++i) sig[i] = sqrtf(fmaxf(M[i][i], 0.f));

    // sort descending, carrying V columns
    #pragma unroll
    for (int i = 0; i < 2; ++i)
        for (int j = i + 1; j < 3; ++j)
            if (sig[j] > sig[i]) {
                float tv = sig[i]; sig[i] = sig[j]; sig[j] = tv;
                #pragma unroll
                for (int k = 0; k < 3; ++k) {
                    float tc = V[k][i]; V[k][i] = V[k][j]; V[k][j] = tc;
                }
            }

    // U columns: u_i = A v_i / sigma_i  (guard rank-2 case on the last column)
    float U[3][3];
    #pragma unroll
    for (int i = 0; i < 3; ++i) {
        float v0 = V[0][i], v1 = V[1][i], v2 = V[2][i];
        float a0 = A[0][0]*v0 + A[0][1]*v1 + A[0][2]*v2;
        float a1 = A[1][0]*v0 + A[1][1]*v1 + A[1][2]*v2;
        float a2 = A[2][0]*v0 + A[2][1]*v1 + A[2][2]*v2;
        float sg = sig[i];
        if (sg > 1e-12f) {
            float inv = 1.0f / sg;
            U[0][i] = a0*inv; U[1][i] = a1*inv; U[2][i] = a2*inv;
        } else if (i == 2) {
            float c0 = U[1][0]*U[2][1] - U[2][0]*U[1][1];
            float c1 = U[2][0]*U[0][1] - U[0][0]*U[2][1];
            float c2 = U[0][0]*U[1][1] - U[1][0]*U[0][1];
            float nn = rsqrtf(fmaxf(c0*c0 + c1*c1 + c2*c2, 1e-30f));
            U[0][2] = c0*nn; U[1][2] = c1*nn; U[2][2] = c2*nn;
        } else {
            U[0][i] = (i == 0) ? 1.f : 0.f;
            U[1][i] = (i == 1) ? 1.f : 0.f;
            U[2][i] = 0.f;
        }
    }

    float d3 = (detA < 0.f) ? -1.f : 1.f;
    float scale = (sig[0] + sig[1] + d3 * sig[2]) / var_s;

    float sR[3][3];
    #pragma unroll
    for (int i = 0; i < 3; ++i)
        #pragma unroll
        for (int j = 0; j < 3; ++j)
            sR[i][j] = scale * (U[i][0]*V[j][0] + U[i][1]*V[j][1] + d3*U[i][2]*V[j][2]);

    float tvec[3];
    #pragma unroll
    for (int i = 0; i < 3; ++i)
        tvec[i] = mud[i] - (sR[i][0]*mus[0] + sR[i][1]*mus[1] + sR[i][2]*mus[2]);

    // Emit WMMA B-operand rows: w[k*4 + d]; k=0..2 -> sR[d][k], k=3 -> t[d]; col 3 = 0
    float* w = wsB + (size_t)b * 16;
    #pragma unroll
    for (int d = 0; d < 3; ++d) {
        w[0*4 + d] = sR[d][0];
        w[1*4 + d] = sR[d][1];
        w[2*4 + d] = sR[d][2];
        w[3*4 + d] = tvec[d];
    }
    #pragma unroll
    for (int k = 0; k < 4; ++k) w[k*4 + 3] = 0.f;
}

// ---------------------------------------------------------------------------
// Kernel 2: HBM-bound affine pass via V_WMMA_F32_16X16X4_F32.
// Block = 256 threads (8 wave32s) handles TILE=2048 pixels of one batch.
// LDS planes padded to PTILE so the two half-waves and the 6 result lanes
// land on disjoint banks; plane 3 holds constant 1.0 so the A-operand load
// is branchless for both half-waves.
// ---------------------------------------------------------------------------
__global__ void __launch_bounds__(256)
transform_kernel(const float* __restrict__ Off,
                 const float* __restrict__ Mean,
                 const float* __restrict__ wsB,
                 float*       __restrict__ Out)
{
    __shared__ __align__(16) float sh_in [4 * PTILE];   // c0,c1,c2,ones
    __shared__ __align__(16) float sh_out[3 * PTILE];

    const int bb   = blockIdx.y;           // batch
    const int tile = blockIdx.x;           // tile within batch
    const int p0   = tile * TILE;          // base flattened pixel

    const int  lane = threadIdx.x & 31;
    const int  wv   = threadIdx.x >> 5;    // wave id 0..7
    const int  n    = lane & 15;
    const bool hi   = lane >= 16;

    // B-operand (4x16 fp32): lanes 0-15 hold rows K=0,1 ; lanes 16-31 rows K=2,3
    const float* w = wsB + (size_t)bb * 16;
    v2f bm;
    bm.x = (n < 4) ? w[(hi ? 2 : 0) * 4 + n] : 0.f;
    bm.y = (n < 4) ? w[(hi ? 3 : 1) * 4 + n] : 0.f;

    // Speculative prefetch of next tile (global_prefetch_b8 path)
    if (tile + 1 < TILES_PER_BATCH) {
        #pragma unroll
        for (int c = 0; c < 3; ++c)
            __builtin_prefetch(Off + ((size_t)bb * 3 + c) * HW + p0 + TILE
                                   + (threadIdx.x & 127) * 16, 0, 0);
    }

    // Stage (Offset*4 + mean) * revert into LDS with coalesced b128 loads
    const float rev[3] = {1.f, -1.f, 1.f};
    for (int i = threadIdx.x; i < 3 * (TILE / 4); i += 256) {
        int c = i / (TILE / 4);
        int j = i % (TILE / 4);
        const float4 o = ((const float4*)(Off  + ((size_t)bb * 3 + c) * HW + p0))[j];
        const float4 m = ((const float4*)(Mean + (size_t)c * HW + p0))[j];
        float r = rev[c];
        float4 v;
        v.x = (o.x * 4.f + m.x) * r;
        v.y = (o.y * 4.f + m.y) * r;
        v.z = (o.z * 4.f + m.z) * r;
        v.w = (o.w * 4.f + m.w) * r;
        ((float4*)&sh_in[c * PTILE])[j] = v;
    }
    // ones plane (A-operand K=3 for the high half-wave)
    {
        float4 one4; one4.x = one4.y = one4.z = one4.w = 1.0f;
        for (int j = threadIdx.x; j < TILE / 4; j += 256)
            ((float4*)&sh_in[3 * PTILE])[j] = one4;
    }
    __syncthreads();

    // Each wave: 256 pixels = 16 WMMAs of 16 pixels each.
    // A (16x4 fp32): lanes 0-15 hold K=0,1 (c0,c1); lanes 16-31 hold K=2,3 (c2,1).
    // Branchless: lane base pointer picks plane pair (0,1) or (2,3).
    const float* pA   = sh_in + (hi ? 2 * PTILE : 0) + n;
    float*       pOut = sh_out + n * PTILE + (hi ? 8 : 0);   // valid when n<3
    const int    wbase = wv * 256;

    #pragma unroll 4
    for (int it = 0; it < 16; ++it) {
        int base = wbase + it * 16;
        v2f a;
        a.x = pA[base];
        a.y = pA[base + PTILE];
        v8f acc = {};
        v8f dres = __builtin_amdgcn_wmma_f32_16x16x4_f32(
            /*neg_a=*/false, a, /*neg_b=*/false, bm,
            /*c_mod=*/(short)0, acc, /*reuse_a=*/false, /*reuse_b=*/false);
        // D layout: lane N=n, VGPR r -> element (M = r + (hi?8:0), N = n).
        // Only N=0..2 are real output channels.
        if (n < 3) {
            float4 lo, hi4;
            lo.x  = dres[0]; lo.y  = dres[1]; lo.z  = dres[2]; lo.w  = dres[3];
            hi4.x = dres[4]; hi4.y = dres[5]; hi4.z = dres[6]; hi4.w = dres[7];
            *((float4*)(pOut + base + 0)) = lo;
            *((float4*)(pOut + base + 4)) = hi4;
        }
    }
    __syncthreads();

    // Coalesced b128 stores to the channel-planar output
    for (int i = threadIdx.x; i < 3 * (TILE / 4); i += 256) {
        int c = i / (TILE / 4);
        int j = i % (TILE / 4);
        ((float4*)(Out + ((size_t)bb * 3 + c) * HW + p0))[j] = ((float4*)&sh_out[c * PTILE])[j];
    }
}

// ---------------------------------------------------------------------------
extern "C" void kernel_launch(void* const* d_in, const int* in_sizes, int n_in,
                              void* d_out, int out_size, void* d_ws, size_t ws_size,
                              hipStream_t stream) {
    const float* Offset = (const float*)d_in[0];   // [64,3,256,256] f32
    const float* Posmap = (const float*)d_in[1];   // [64,3,256,256] f32
    const float* Meanp  = (const float*)d_in[2];   // [3,256,256]    f32
    const int*   uv     = (const int*)  d_in[3];   // [68,2]         i32
    float*       out    = (float*)d_out;           // [64,3,256,256] f32
    float*       wsB    = (float*)d_ws;            // 64*16 floats

    umeyama_kernel<<<2, 32, 0, stream>>>(Offset, Posmap, Meanp, uv, wsB);

    dim3 grid(TILES_PER_BATCH, BATCH);
    transform_kernel<<<grid, 256, 0, stream>>>(Offset, Meanp, wsB, out);
}